// ReductionLayerSVD_62448824484600
// MI455X (gfx1250) — compile-verified
//
#include <hip/hip_runtime.h>
#include <hip/hip_bf16.h>

// Rank-8 truncated-SVD reconstruction via block subspace iteration:
//   V <- orth( A^T (A V) ) repeated; Out = (A V) V^T  ( == U_r S_r V_r^T ).
// Heavy GEMMs use V_WMMA_F32_16X16X4_F32; V-panel staging uses the CDNA5
// async global->LDS DMA path (GLOBAL_LOAD_ASYNC_TO_LDS_B128 + s_wait_asynccnt).

typedef __attribute__((ext_vector_type(2))) float v2f;
typedef __attribute__((ext_vector_type(8))) float v8f;
typedef __attribute__((address_space(3))) float lds_f;

#define NB   128
#define NN   512
#define RNK  8
#define PAD  16          // WMMA N dimension (cols 8..15 are zero)
#define ITERS 20

__device__ __forceinline__ v8f wmma_f32(v2f a, v2f b, v8f c) {
    // 8 args: (neg_a, A, neg_b, B, c_mod, C, reuse_a, reuse_b)
    return __builtin_amdgcn_wmma_f32_16x16x4_f32(false, a, false, b, (short)0, c,
                                                 false, false);
}

// Stage the 16 KB V panel (512x8 f32) global -> LDS with async DMA.
// Each of 256 threads issues 4 x B128 transfers; completion via ASYNCcnt.
__device__ __forceinline__ void load_V_async(float* Vs, const float* __restrict__ Vb,
                                             int tid) {
    #pragma unroll
    for (int p = 0; p < 4; ++p) {
        const int e = (p * 256 + tid) * 4;           // float index, 16B granules
        lds_f* dst = (lds_f*)(Vs + e);               // 32-bit LDS byte offset
        const float* src = Vb + e;
        asm volatile("global_load_async_to_lds_b128 %0, %1, off"
                     :: "v"(dst), "v"(src) : "memory");
    }
    asm volatile("s_wait_asynccnt 0x0" ::: "memory");
}

// ---------------------------------------------------------------------------
// Kernel 1: per-batch subspace iteration. One workgroup (8 waves) per batch.
//   Xs: 512x16 candidate basis in LDS (cols 8..15 == 0)
//   Ys: 512x16 intermediate A*X in LDS (also reused as reduction scratch)
// ---------------------------------------------------------------------------
__global__ void __launch_bounds__(256)
svd_power_kernel(const float* __restrict__ A, float* __restrict__ Vg) {
    __shared__ float smem[2 * NN * PAD];          // 64 KB
    float* Xs = smem;
    float* Ys = smem + NN * PAD;

    const int b    = blockIdx.x;
    const int tid  = threadIdx.x;
    const int lane = tid & 31;
    const int wave = tid >> 5;
    const float* __restrict__ Ab = A + (size_t)b * NN * NN;

    // Deterministic pseudo-random init of X (full rank w.h.p.), zero padding cols.
    for (int idx = tid; idx < NN * PAD; idx += 256) {
        int r = idx >> 4, j = idx & 15;
        float v = 0.0f;
        if (j < RNK) {
            unsigned s = (unsigned)((b * NN + r) * 8 + j) * 2654435761u + 12345u;
            s ^= s >> 13; s *= 2246822519u; s ^= s >> 16;
            v = (float)(s & 0xFFFFFFu) * (1.0f / 8388608.0f) - 1.0f;
        }
        Xs[idx] = v;
    }
    __syncthreads();

    const int m    = lane & 15;        // M index within tile (also N index for B/C)
    const int kOff = (lane >> 4) << 1; // lanes 0-15 -> K{0,1}, lanes 16-31 -> K{2,3}

    for (int it = 0; it < ITERS; ++it) {
        // ---- GEMM1: Ys = A * Xs  (each wave owns 64 rows = 4 MxK tiles) ----
        for (int t = 0; t < 4; ++t) {
            const int row0 = wave * 64 + t * 16;
            const float* Arow = Ab + (size_t)(row0 + m) * NN;
            v8f acc = {};
            for (int k = 0; k < NN; k += 4) {
                if ((k & 31) == 0)                      // one per 128B line, near scope
                    __builtin_prefetch(Arow + k + 256, 0, 3);
                v2f a = *(const v2f*)(Arow + k + kOff);
                v2f bb;
                bb.x = Xs[(k + kOff + 0) * PAD + m];
                bb.y = Xs[(k + kOff + 1) * PAD + m];
                acc = wmma_f32(a, bb, acc);
            }
            const int rb = row0 + ((lane >> 4) << 3);
            #pragma unroll
            for (int r = 0; r < 8; ++r) Ys[(rb + r) * PAD + m] = acc[r];
        }
        __syncthreads();

        // ---- GEMM2: Xs = A^T * Ys  (each wave owns 64 columns of A) ----
        for (int t = 0; t < 4; ++t) {
            const int c0 = wave * 64 + t * 16;
            v8f acc = {};
            for (int k = 0; k < NN; k += 4) {
                const int kk = k + kOff;
                v2f a, bb;
                a.x = Ab[(size_t)(kk + 0) * NN + c0 + m];
                a.y = Ab[(size_t)(kk + 1) * NN + c0 + m];
                bb.x = Ys[(kk + 0) * PAD + m];
                bb.y = Ys[(kk + 1) * PAD + m];
                acc = wmma_f32(a, bb, acc);
            }
            const int rb = c0 + ((lane >> 4) << 3);
            #pragma unroll
            for (int r = 0; r < 8; ++r) Xs[(rb + r) * PAD + m] = acc[r];
        }
        __syncthreads();

        // ---- Modified Gram-Schmidt on Xs cols 0..7 (Ys reused as scratch) ----
        float* red = Ys;
        for (int j = 0; j < RNK; ++j) {
            for (int i = 0; i < j; ++i) {
                float p = Xs[tid * PAD + i] * Xs[tid * PAD + j]
                        + Xs[(tid + 256) * PAD + i] * Xs[(tid + 256) * PAD + j];
                red[tid] = p;
                __syncthreads();
                for (int s = 128; s > 0; s >>= 1) {
                    if (tid < s) red[tid] += red[tid + s];
                    __syncthreads();
                }
                float d = red[0];
                __syncthreads();
                Xs[tid * PAD + j]         -= d * Xs[tid * PAD + i];
                Xs[(tid + 256) * PAD + j] -= d * Xs[(tid + 256) * PAD + i];
                __syncthreads();
            }
            float p = Xs[tid * PAD + j] * Xs[tid * PAD + j]
                    + Xs[(tid + 256) * PAD + j] * Xs[(tid + 256) * PAD + j];
            red[tid] = p;
            __syncthreads();
            for (int s = 128; s > 0; s >>= 1) {
                if (tid < s) red[tid] += red[tid + s];
                __syncthreads();
            }
            float inv = rsqrtf(red[0] + 1e-20f);
            __syncthreads();
            Xs[tid * PAD + j]         *= inv;
            Xs[(tid + 256) * PAD + j] *= inv;
            __syncthreads();
        }
    }

    // Write orthonormal V (512x8, row-major) to workspace.
    for (int idx = tid; idx < NN * RNK; idx += 256) {
        int r = idx >> 3, j = idx & 7;
        Vg[(size_t)b * NN * RNK + idx] = Xs[r * PAD + j];
    }
}

// ---------------------------------------------------------------------------
// Kernel 2: W = A * V   (512x8 per batch; cheap, VALU dot products)
// ---------------------------------------------------------------------------
__global__ void __launch_bounds__(256)
svd_w_kernel(const float* __restrict__ A, const float* __restrict__ Vg,
             float* __restrict__ Wg) {
    __shared__ float Vs[NN * RNK];                 // 16 KB
    const int b = blockIdx.x, tid = threadIdx.x;
    const float* __restrict__ Ab = A + (size_t)b * NN * NN;
    load_V_async(Vs, Vg + (size_t)b * NN * RNK, tid);
    __syncthreads();

    #pragma unroll
    for (int rr = 0; rr < 2; ++rr) {
        const int r = tid * 2 + rr;
        const float* Ar = Ab + (size_t)r * NN;
        float acc[RNK] = {};
        for (int k = 0; k < NN; k += 4) {
            float4 av = *(const float4*)(Ar + k);
            #pragma unroll
            for (int j = 0; j < RNK; ++j) {
                acc[j] = fmaf(av.x, Vs[(k + 0) * RNK + j], acc[j]);
                acc[j] = fmaf(av.y, Vs[(k + 1) * RNK + j], acc[j]);
                acc[j] = fmaf(av.z, Vs[(k + 2) * RNK + j], acc[j]);
                acc[j] = fmaf(av.w, Vs[(k + 3) * RNK + j], acc[j]);
            }
        }
        #pragma unroll
        for (int j = 0; j < RNK; ++j)
            Wg[(size_t)b * NN * RNK + (size_t)r * RNK + j] = acc[j];
    }
}

// ---------------------------------------------------------------------------
// Kernel 3: Out = W * V^T  (rank-8 GEMM via two K=4 f32 WMMAs per 16x16 tile).
// Grid: (4 blocks, 128 batches); each wave owns one 16-row u-tile, loops v.
// ---------------------------------------------------------------------------
__global__ void __launch_bounds__(256)
svd_outer_kernel(const float* __restrict__ Wg, const float* __restrict__ Vg,
                 float* __restrict__ Out) {
    __shared__ float Vs[NN * RNK];                 // 16 KB
    const int b = blockIdx.y, blk = blockIdx.x;
    const int tid = threadIdx.x, lane = tid & 31, wave = tid >> 5;
    const float* __restrict__ Wb = Wg + (size_t)b * NN * RNK;
    load_V_async(Vs, Vg + (size_t)b * NN * RNK, tid);
    __syncthreads();

    const int m    = lane & 15;
    const int kOff = (lane >> 4) << 1;
    const int u0   = (blk * 8 + wave) * 16;

    // A fragments (W tile rows, K = 0..3 and 4..7) held in registers.
    const float* Wr = Wb + (size_t)(u0 + m) * RNK;
    v2f a0, a1;
    a0.x = Wr[kOff + 0]; a0.y = Wr[kOff + 1];
    a1.x = Wr[kOff + 4]; a1.y = Wr[kOff + 5];

    float* __restrict__ Ob = Out + (size_t)b * NN * NN;
    for (int v0 = 0; v0 < NN; v0 += 16) {
        const float* Vr = Vs + (v0 + m) * RNK;     // B[k][n] = V[v0+n][k]
        v2f b0, b1;
        b0.x = Vr[kOff + 0]; b0.y = Vr[kOff + 1];
        b1.x = Vr[kOff + 4]; b1.y = Vr[kOff + 5];
        v8f acc = {};
        acc = wmma_f32(a0, b0, acc);
        acc = wmma_f32(a1, b1, acc);
        const int ub = u0 + ((lane >> 4) << 3);
        #pragma unroll
        for (int r = 0; r < 8; ++r)
            Ob[(size_t)(ub + r) * NN + v0 + m] = acc[r];
    }
}

// ---------------------------------------------------------------------------
extern "C" void kernel_launch(void* const* d_in, const int* in_sizes, int n_in,
                              void* d_out, int out_size, void* d_ws, size_t ws_size,
                              hipStream_t stream) {
    (void)in_sizes; (void)n_in; (void)out_size; (void)ws_size;
    const float* A  = (const float*)d_in[0];
    float* Out      = (float*)d_out;
    float* Vg       = (float*)d_ws;                    // 128*512*8 f32 = 2 MB
    float* Wg       = Vg + (size_t)NB * NN * RNK;      // another 2 MB

    svd_power_kernel<<<dim3(NB), dim3(256), 0, stream>>>(A, Vg);
    svd_w_kernel    <<<dim3(NB), dim3(256), 0, stream>>>(A, Vg, Wg);
    svd_outer_kernel<<<dim3(4, NB), dim3(256), 0, stream>>>(Wg, Vg, Out);
}